// mParametricLIF_523986010151
// MI455X (gfx1250) — compile-verified
//
#include <hip/hip_runtime.h>
#include <cstdint>
#include <cstddef>

// ---------------------------------------------------------------------------
// Parametric LIF forward scan, CDNA5 (gfx1250).  x: [B=64, T=64, N=8192] f32.
//
//   u    = last*sig + x[b,t,n];  s = (u >= th);
//   last = (u + lamb*(u-last)) * (1-s);     out[b,t,n] = s
//
// Bandwidth-bound: 256 MB total traffic -> ~11us floor @ 23.3 TB/s.
// Each thread owns 4 consecutive channels (one b128 per lane, 512B/wave/step,
// fully coalesced). Input is staged via the CDNA5 async path
// (global_load_async_to_lds_b128, GVS saddr form) in an 8-deep per-wave
// pipeline throttled by s_wait_asynccnt; stores are nontemporal and loads
// carry th:TH_LOAD_NT since the stream exceeds the 192MB L2.
// ---------------------------------------------------------------------------

typedef __attribute__((ext_vector_type(4))) float floatx4;

constexpr int LIF_T   = 64;
constexpr int LIF_N   = 8192;
constexpr int BLOCK   = 256;
constexpr int WAVE    = 32;                    // gfx1250 is wave32-only
constexpr int PF      = 8;                     // async pipeline depth (pow2)
constexpr int WPB     = BLOCK / WAVE;          // 8 waves / block
constexpr int BUF_STRIDE = WAVE * 16;          // 512 B per buffer per wave
constexpr int WAVE_BUF   = PF * BUF_STRIDE;    // 4 KB per wave
constexpr int SMEM_BYTES = WPB * WAVE_BUF;     // 32 KB per block
constexpr int TSTRIDE_B  = LIF_N * 4;          // 32 KB byte stride per timestep
constexpr int BLOCKS_PER_ROW = (LIF_N / 4) / BLOCK;   // 8

__device__ __forceinline__
floatx4 lif_step(floatx4 xv, floatx4& last, float sig, float lam, float th)
{
    floatx4 s;
#pragma unroll
    for (int i = 0; i < 4; ++i) {
        float u  = fmaf(last[i], sig, xv[i]);
        bool  f  = (u >= th);
        s[i]     = f ? 1.0f : 0.0f;
        // last = (u + lam*(u-last)) * (1-s) == s ? 0 : fma(lam, u-last, u)
        last[i]  = f ? 0.0f : fmaf(lam, u - last[i], u);
    }
    return s;
}

__global__ __launch_bounds__(BLOCK)
void lif_scan_async_kernel(const float* __restrict__ x,
                           const float* __restrict__ tau_p,
                           const float* __restrict__ lamb_p,
                           const float* __restrict__ th_p,
                           float* __restrict__ out)
{
    __shared__ __align__(16) unsigned char smem[SMEM_BYTES];

    const int tid  = (int)threadIdx.x;
    const int wave = tid >> 5;
    const int lane = tid & (WAVE - 1);

    const int b  = (int)blockIdx.y;       // batch row
    const int nb = (int)blockIdx.x;       // chunk within the N dimension

    // element offset of this block's chunk start (lane part handled below)
    const size_t row = (size_t)b * LIF_T * LIF_N + (size_t)nb * (BLOCK * 4);

    // GVS addressing: uniform 64-bit base in SGPRs, loop-invariant per-lane
    // 32-bit byte offset in one VGPR.
    const uint64_t gsrc_base = (uint64_t)(uintptr_t)(x + row);
    const uint32_t voff      = (uint32_t)tid * 16u;
    float*         gdst      = out + row + (size_t)tid * 4;

    // LDS slot byte addresses for this lane (low 32 bits of a generic LDS
    // pointer == LDS byte offset on gfx1250)
    unsigned char* mybuf = &smem[wave * WAVE_BUF + lane * 16];
    uint32_t lds_addr[PF];
#pragma unroll
    for (int k = 0; k < PF; ++k)
        lds_addr[k] = (uint32_t)(uintptr_t)(mybuf + k * BUF_STRIDE);

    // scalar parameters
    const float sig = 1.0f / (1.0f + expf(-tau_p[0]));
    const float lam = lamb_p[0];
    const float th  = th_p[0];

    // ---- prologue: fill the async pipeline ---------------------------------
#pragma unroll
    for (int k = 0; k < PF; ++k) {
        uint64_t sa = gsrc_base + (uint64_t)k * TSTRIDE_B;
        asm volatile("global_load_async_to_lds_b128 %0, %1, %2 th:TH_LOAD_NT"
                     :: "v"(lds_addr[k]), "v"(voff), "s"(sa)
                     : "memory");
    }

    floatx4 last = {0.0f, 0.0f, 0.0f, 0.0f};

    // ---- main loop: consume t, refill slot with t+PF -----------------------
#pragma unroll 4
    for (int t = 0; t < LIF_T - PF; ++t) {
        const int k = t & (PF - 1);

        // async loads complete in order: <=PF-1 outstanding => buffer t landed
        asm volatile("s_wait_asynccnt 7" ::: "memory");

        const floatx4 xv = *(const floatx4*)(mybuf + k * BUF_STRIDE);
        const floatx4 sv = lif_step(xv, last, sig, lam, th);
        __builtin_nontemporal_store(sv, (floatx4*)(gdst + (size_t)t * LIF_N));

        // refill slot k with timestep t+PF; s_wait_dscnt 0 ensures the
        // ds_load above drained before the async engine overwrites the buffer
        uint64_t sa = gsrc_base + (uint64_t)(t + PF) * TSTRIDE_B;
        asm volatile("s_wait_dscnt 0\n\t"
                     "global_load_async_to_lds_b128 %0, %1, %2 th:TH_LOAD_NT"
                     :: "v"(lds_addr[k]), "v"(voff), "s"(sa)
                     : "memory");
    }

    // ---- tail: drain the pipeline ------------------------------------------
    asm volatile("s_wait_asynccnt 0" ::: "memory");
#pragma unroll
    for (int t = LIF_T - PF; t < LIF_T; ++t) {
        const int k = t & (PF - 1);
        const floatx4 xv = *(const floatx4*)(mybuf + k * BUF_STRIDE);
        const floatx4 sv = lif_step(xv, last, sig, lam, th);
        __builtin_nontemporal_store(sv, (floatx4*)(gdst + (size_t)t * LIF_N));
    }
}

extern "C" void kernel_launch(void* const* d_in, const int* in_sizes, int n_in,
                              void* d_out, int out_size, void* d_ws, size_t ws_size,
                              hipStream_t stream)
{
    (void)n_in; (void)out_size; (void)d_ws; (void)ws_size;

    const float* x     = (const float*)d_in[0];
    const float* tau_p = (const float*)d_in[1];
    const float* lamb  = (const float*)d_in[2];
    const float* th    = (const float*)d_in[3];
    float*       out   = (float*)d_out;

    const int B = in_sizes[0] / (LIF_T * LIF_N);    // reference: 64

    dim3 grid((unsigned)BLOCKS_PER_ROW, (unsigned)B);   // (8, 64)
    dim3 block((unsigned)BLOCK);

    lif_scan_async_kernel<<<grid, block, 0, stream>>>(x, tau_p, lamb, th, out);
}